// GCN3DDecoder_13554916786448
// MI455X (gfx1250) — compile-verified
//
#include <hip/hip_runtime.h>
#include <math.h>

typedef __attribute__((ext_vector_type(2))) float v2f;
typedef __attribute__((ext_vector_type(8))) float v8f;

#define BS        4
#define NVERT     128
#define KTOT      332        // 128 + 128 + 64 + 12
#define KPAD      336        // 21 tiles of 16
#define NKTILE    21

// ---------------------------------------------------------------------------
// K0: fm0 = feature_global @ W1 + b1   (4 x 512) @ (512 x 1024)
// Stored flat as P[b*1024 + v*8 + j]  (vertex signature rows contiguous).
// ---------------------------------------------------------------------------
__global__ void k0_fm0(const float* __restrict__ fg, const float* __restrict__ W1,
                       const float* __restrict__ b1, float* __restrict__ P) {
    int t = blockIdx.x * blockDim.x + threadIdx.x;      // 4096 threads
    int b = t >> 10;
    int d = t & 1023;
    float acc = b1[d];
    const float* f = fg + b * 512;
    for (int i = 0; i < 512; ++i)
        acc = fmaf(f[i], W1[i * 1024 + d], acc);
    P[t] = acc;
}

// ---------------------------------------------------------------------------
// K1: supB[j][k] = sum over 128-d block j of normalize(dirs[:,k]) , j<8
// Concatenated column space: [dirs1 | dirst1 | dirst2 | dirst3], padded to 336.
// ---------------------------------------------------------------------------
__global__ void k1_supB(const float* __restrict__ dirs1, const float* __restrict__ dirst1,
                        const float* __restrict__ dirst2, const float* __restrict__ dirst3,
                        float* __restrict__ supB) {
    int k = blockIdx.x * blockDim.x + threadIdx.x;
    if (k >= KPAD) return;
    const float* src = nullptr; int ldk = 0, col = 0;
    if      (k < 128) { src = dirs1;  ldk = 128; col = k;       }
    else if (k < 256) { src = dirst1; ldk = 128; col = k - 128; }
    else if (k < 320) { src = dirst2; ldk = 64;  col = k - 256; }
    else if (k < 332) { src = dirst3; ldk = 12;  col = k - 320; }
    if (!src) {                       // pad columns -> zero (theta=0, harmless under relu/max)
        for (int j = 0; j < 8; ++j) supB[j * KPAD + k] = 0.0f;
        return;
    }
    float ss = 0.0f;
    float blk[8] = {0,0,0,0,0,0,0,0};
    for (int d = 0; d < 1024; ++d) {
        float x = src[d * ldk + col];
        ss = fmaf(x, x, ss);
        blk[d >> 7] += x;
    }
    float rn = 1.0f / fmaxf(sqrtf(ss), 1e-12f);         // normalize(dirs, axis=0), EPS=1e-12
    for (int j = 0; j < 8; ++j) supB[j * KPAD + k] = blk[j] * rn;
}

// ---------------------------------------------------------------------------
// K2: for each (b, v, ktile): Mx[b,v,16k..] = max_w relu( dhat8(v,w) . supB )
// One wave per (b,v,ktile); M=128 (w) swept as 8 tiles of 16, K=8 via two
// V_WMMA_F32_16X16X4_F32 per tile (fully unrolled -> 16 static WMMAs).
// Self (w==v) gives theta=0 which cannot affect the nonnegative max.
// ---------------------------------------------------------------------------
__global__ __launch_bounds__(256) void k2_theta_max(const float* __restrict__ P,
                                                    const float* __restrict__ supB,
                                                    float* __restrict__ Mx) {
    const int  lane = threadIdx.x & 31;
    const int  wid  = (blockIdx.x * blockDim.x + threadIdx.x) >> 5;
    const int  kt   = wid % NKTILE;
    const int  v    = (wid / NKTILE) & (NVERT - 1);
    const int  b    = wid / (NKTILE * NVERT);
    const bool hh   = (lane & 16) != 0;   // lane half: 0 -> lanes 0-15, 1 -> lanes 16-31
    const int  n16  = lane & 15;
    const int  col  = kt * 16 + n16;

    // B tile (4x16 f32): lane n = N, VGPR0 holds K=0/2, VGPR1 holds K=1/3 (by half)
    const int kb = hh ? 2 : 0;
    v2f B01, B23;
    B01[0] = supB[(kb + 0) * KPAD + col];
    B01[1] = supB[(kb + 1) * KPAD + col];
    B23[0] = supB[(kb + 4) * KPAD + col];
    B23[1] = supB[(kb + 5) * KPAD + col];

    // center vertex signature (8 floats, 32B aligned)
    const float4* pv4 = reinterpret_cast<const float4*>(P + ((size_t)b * NVERT + v) * 8);
    float4 cl = pv4[0], ch = pv4[1];
    const float pv0 = cl.x, pv1 = cl.y, pv2 = cl.z, pv3 = cl.w;
    const float pv4f = ch.x, pv5 = ch.y, pv6 = ch.z, pv7 = ch.w;

    v8f vmax = {};                                     // implicit relu floor = 0
#pragma unroll
    for (int wt = 0; wt < 8; ++wt) {
        int w = wt * 16 + n16;                         // A row M = n16 (both halves)
        const float4* pw4 = reinterpret_cast<const float4*>(P + ((size_t)b * NVERT + w) * 8);
        float4 lo = pw4[0], hi = pw4[1];
        float d0 = lo.x - pv0, d1 = lo.y - pv1, d2 = lo.z - pv2, d3 = lo.w - pv3;
        float d4 = hi.x - pv4f, d5 = hi.y - pv5, d6 = hi.z - pv6, d7 = hi.w - pv7;
        float ss = d0*d0 + d1*d1 + d2*d2 + d3*d3 + d4*d4 + d5*d5 + d6*d6 + d7*d7;
        // ||diff_1024|| = sqrt(128 * ||diff_8||^2); EPS = 1e-12; w==v -> dh = 0
        float rn = 1.0f / fmaxf(sqrtf(ss * 128.0f), 1e-12f);

        // A tile (16x4 f32): lane M = n16, VGPR0 K=0/2, VGPR1 K=1/3 (by half).
        // Explicit selects (one v_cndmask each) instead of dynamic array indexing.
        v2f A01, A23;
        A01[0] = (hh ? d2 : d0) * rn;
        A01[1] = (hh ? d3 : d1) * rn;
        A23[0] = (hh ? d6 : d4) * rn;
        A23[1] = (hh ? d7 : d5) * rn;

        v8f c = {};
        c = __builtin_amdgcn_wmma_f32_16x16x4_f32(false, A01, false, B01, (short)0, c, false, false);
        c = __builtin_amdgcn_wmma_f32_16x16x4_f32(false, A23, false, B23, (short)0, c, false, false);
#pragma unroll
        for (int r = 0; r < 8; ++r) vmax[r] = fmaxf(vmax[r], c[r]);
    }
    // reduce over the 16 rows of the accumulated tile: 8 VGPRs + lane<->lane+16
    float m = vmax[0];
#pragma unroll
    for (int r = 1; r < 8; ++r) m = fmaxf(m, vmax[r]);
    // ds_swizzle 0x7C10 (and=0x10, or=0, xor=0x1f): lanes 0-15 read lane 31, 16-31 read lane 15
    float ms = __int_as_float(__builtin_amdgcn_ds_swizzle(__float_as_int(m), 0x7C10));
    m = fmaxf(m, ms);
    if (lane < 16)
        Mx[((size_t)b * NVERT + v) * KPAD + col] = m;
}

// ---------------------------------------------------------------------------
// K3: per-vertex head: fm1 -> conv_transpose(32) -> conv_transpose(16) -> tanh(3)
// Segments in Mx row: [0,128)=surface, [128,256)=t1, [256,320)=t2, [320,332)=t3
// ---------------------------------------------------------------------------
__global__ void k3_head(const float* __restrict__ Mx,
                        const float* __restrict__ Wt1, const float* __restrict__ bt1,
                        const float* __restrict__ Wt2, const float* __restrict__ bt2,
                        const float* __restrict__ Wt3, const float* __restrict__ bt3,
                        float* __restrict__ out) {
    int t = blockIdx.x * blockDim.x + threadIdx.x;     // 512 = BS*NVERT
    if (t >= BS * NVERT) return;
    const float* mx = Mx + (size_t)t * KPAD;

    float fm1[32];
    for (int c = 0; c < 32; ++c)
        fm1[c] = fmaxf(mx[c] + mx[32 + c] + mx[64 + c] + mx[96 + c], 0.0f);

    float fm2[32];
    for (int c = 0; c < 32; ++c) {
        float acc = bt1[c];
        float dc = 0.0f;
        for (int i = 0; i < 32; ++i) dc = fmaf(fm1[i], Wt1[i * 160 + c], dc);
        acc += dc;
        for (int s = 0; s < 4; ++s) {
            float th = mx[128 + s * 32 + c];
            int colw = 32 + s * 32 + c;
            float dp = 0.0f;
            for (int i = 0; i < 32; ++i) dp = fmaf(fm1[i], Wt1[i * 160 + colw], dp);
            acc = fmaf(th, dp, acc);
        }
        fm2[c] = fmaxf(acc, 0.0f);
    }

    float fm4[16];
    for (int c = 0; c < 16; ++c) {
        float acc = bt2[c];
        float dc = 0.0f;
        for (int i = 0; i < 32; ++i) dc = fmaf(fm2[i], Wt2[i * 80 + c], dc);
        acc += dc;
        for (int s = 0; s < 4; ++s) {
            float th = mx[256 + s * 16 + c];
            int colw = 16 + s * 16 + c;
            float dp = 0.0f;
            for (int i = 0; i < 32; ++i) dp = fmaf(fm2[i], Wt2[i * 80 + colw], dp);
            acc = fmaf(th, dp, acc);
        }
        fm4[c] = fmaxf(acc, 0.0f);
    }

    for (int c = 0; c < 3; ++c) {
        float acc = bt3[c];
        float dc = 0.0f;
        for (int i = 0; i < 16; ++i) dc = fmaf(fm4[i], Wt3[i * 15 + c], dc);
        acc += dc;
        for (int s = 0; s < 4; ++s) {
            float th = mx[320 + s * 3 + c];
            int colw = 3 + s * 3 + c;
            float dp = 0.0f;
            for (int i = 0; i < 16; ++i) dp = fmaf(fm4[i], Wt3[i * 15 + colw], dp);
            acc = fmaf(th, dp, acc);
        }
        out[t * 3 + c] = tanhf(acc);
    }
}

// ---------------------------------------------------------------------------
extern "C" void kernel_launch(void* const* d_in, const int* in_sizes, int n_in,
                              void* d_out, int out_size, void* d_ws, size_t ws_size,
                              hipStream_t stream) {
    (void)in_sizes; (void)n_in; (void)out_size; (void)ws_size;
    const float* fg     = (const float*)d_in[0];
    const float* W1     = (const float*)d_in[1];
    const float* b1     = (const float*)d_in[2];
    const float* dirs1  = (const float*)d_in[3];
    const float* Wt1    = (const float*)d_in[4];
    const float* bt1    = (const float*)d_in[5];
    const float* dirst1 = (const float*)d_in[6];
    const float* Wt2    = (const float*)d_in[7];
    const float* bt2    = (const float*)d_in[8];
    const float* dirst2 = (const float*)d_in[9];
    const float* Wt3    = (const float*)d_in[10];
    const float* bt3    = (const float*)d_in[11];
    const float* dirst3 = (const float*)d_in[12];
    float* out = (float*)d_out;

    float* ws   = (float*)d_ws;
    float* P    = ws;                 // 4*1024            = 4096 floats
    float* supB = ws + 4096;          // 8*336             = 2688 floats
    float* Mx   = ws + 6784;          // 4*128*336         = 172032 floats (~700 KB total)

    k0_fm0<<<dim3(16), dim3(256), 0, stream>>>(fg, W1, b1, P);
    k1_supB<<<dim3(6), dim3(64), 0, stream>>>(dirs1, dirst1, dirst2, dirst3, supB);
    // 4 * 128 * 21 = 10752 waves -> 1344 blocks of 8 waves (exact cover, EXEC all-ones)
    k2_theta_max<<<dim3(1344), dim3(256), 0, stream>>>(P, supB, Mx);
    k3_head<<<dim3(2), dim3(256), 0, stream>>>(Mx, Wt1, bt1, Wt2, bt2, Wt3, bt3, out);
}